// Wide_72404558676679
// MI455X (gfx1250) — compile-verified
//
#include <hip/hip_runtime.h>

// Wide hash-bucket embedding pooling for MI455X (gfx1250).
//   bucket = idx % 4e6 ; x = table[bucket] * value ; segment-sum over 64 fields per row.
// Memory-bound streaming kernel (~102 MB @ 23.3 TB/s ~ 4.5 us floor).
// CDNA5 paths used: GLOBAL_LOAD_ASYNC_TO_LDS_B32 (+ s_wait_asynccnt) for input
// staging, ds_add_f32 LDS atomics for per-wave segment accumulation.

#define BATCH        16384
#define SEQ          500
#define NUM_BUCKETS  4000000u
#define NSEG         64        // fields 0..63 (field 0 dropped in output)
#define NOUT         63
#define BLOCK        256
#define NWAVE        (BLOCK / 32)
#define SEQ_PAD      512

// Generic (flat) pointer to an LDS object: low 32 bits are the LDS byte offset
// (flat->LDS aperture mapping truncates to addr[31:0]).
__device__ __forceinline__ unsigned lds_offset(const void* p) {
  return (unsigned)(unsigned long long)p;
}

// CDNA5 async global -> LDS copy, 4 bytes per active lane.
//   VDST = per-lane VGPR holding LDS byte offset
//   VADDR = per-lane VGPR holding unsigned byte offset from SGPR base (GVS mode)
//   SADDR = 64-bit SGPR base address
__device__ __forceinline__ void async_load_b32_to_lds(unsigned lds_off,
                                                      unsigned byte_off,
                                                      const void* base) {
  asm volatile("global_load_async_to_lds_b32 %0, %1, %2 offset:0"
               :
               : "v"(lds_off), "v"(byte_off), "s"(base)
               : "memory");
}

__device__ __forceinline__ void wait_asynccnt0() {
  asm volatile("s_wait_asynccnt 0" ::: "memory");
}

__global__ __launch_bounds__(BLOCK) void wide_pool_kernel(
    const int*   __restrict__ indexes,   // [BATCH*SEQ]
    const int*   __restrict__ fields,    // [BATCH*SEQ]
    const float* __restrict__ values,    // [BATCH*SEQ]
    const float* __restrict__ table,     // [NUM_BUCKETS] (dim-1 embedding)
    float*       __restrict__ out)       // [BATCH*NOUT]
{
  __shared__ int   s_idx[SEQ_PAD];
  __shared__ int   s_fld[SEQ_PAD];
  __shared__ float s_val[SEQ_PAD];
  __shared__ float s_acc[NWAVE * NSEG];  // per-wave segment accumulators

  const int      row  = blockIdx.x;
  const int      t    = threadIdx.x;
  const int      wave = t >> 5;
  const unsigned gofs = ((unsigned)row * SEQ + (unsigned)t) * 4u;

  // ---- kick off async staging of this row's inputs into LDS ----
  async_load_b32_to_lds(lds_offset(&s_idx[t]), gofs, indexes);
  async_load_b32_to_lds(lds_offset(&s_fld[t]), gofs, fields);
  async_load_b32_to_lds(lds_offset(&s_val[t]), gofs, values);
  if (t < SEQ - BLOCK) {  // tail chunk: elements 256..499
    const unsigned gofs2 = gofs + (unsigned)BLOCK * 4u;
    async_load_b32_to_lds(lds_offset(&s_idx[t + BLOCK]), gofs2, indexes);
    async_load_b32_to_lds(lds_offset(&s_fld[t + BLOCK]), gofs2, fields);
    async_load_b32_to_lds(lds_offset(&s_val[t + BLOCK]), gofs2, values);
  }

  // zero accumulators while the async copies are in flight
  s_acc[t]         = 0.0f;
  s_acc[t + BLOCK] = 0.0f;   // NWAVE*NSEG == 2*BLOCK

  wait_asynccnt0();          // per-wave: our lanes' staged data has landed
  __syncthreads();           // accumulators zeroed + staging visible block-wide

  // ---- gather + weighted per-wave segment accumulation ----
#pragma unroll
  for (int it = 0; it < 2; ++it) {
    const int k = t + it * BLOCK;
    if (k < SEQ) {
      const unsigned bucket = (unsigned)s_idx[k] % NUM_BUCKETS;
      const int      f      = s_fld[k] & (NSEG - 1);
      float          e      = table[bucket];           // random 4B gather, L2-resident
      if (bucket == 0u) e = 0.0f;                      // padding_idx = 0
      const float    x      = e * s_val[k];
      atomicAdd(&s_acc[wave * NSEG + f], x);           // ds_add_f32, wave-private bank
    }
  }

  __syncthreads();

  // ---- reduce the per-wave banks, drop field 0, write [row, 0..62] ----
  if (t < NOUT) {
    const int f = t + 1;
    float sum = 0.0f;
#pragma unroll
    for (int w = 0; w < NWAVE; ++w) sum += s_acc[w * NSEG + f];
    out[row * NOUT + t] = sum;
  }
}

extern "C" void kernel_launch(void* const* d_in, const int* in_sizes, int n_in,
                              void* d_out, int out_size, void* d_ws, size_t ws_size,
                              hipStream_t stream) {
  const int*   indexes = (const int*)d_in[0];
  const int*   fields  = (const int*)d_in[1];
  const float* values  = (const float*)d_in[2];
  const float* table   = (const float*)d_in[3];
  float*       out     = (float*)d_out;

  wide_pool_kernel<<<BATCH, BLOCK, 0, stream>>>(indexes, fields, values, table, out);
}